// RoIAlign_326417515284
// MI455X (gfx1250) — compile-verified
//
#include <hip/hip_runtime.h>

// RoI Align for MI455X (gfx1250, wave32).
// Memory-bound: 411 MB NT stores + L2-resident 64 MB features.
// Bilinear combine is done exactly on the matrix pipe via chained
// V_WMMA_F32_16X16X4_F32 (K = 4 bilinear corners).

typedef float v2f __attribute__((ext_vector_type(2)));
typedef float v8f __attribute__((ext_vector_type(8)));

#define SPATIAL_SCALE 0.0625f
#define C_  1024
#define H_  64
#define W_  64
#define HW_ (H_ * W_)
#define NBINS 49        // 7x7
#define CH_TILES (C_ / 16)   // 64
#define BIN_GROUPS 4         // ceil(49/16)
#define WAVES_PER_ROI (CH_TILES * BIN_GROUPS)  // 256

__global__ __launch_bounds__(256) void roialign_wmma_kernel(
    const float* __restrict__ feat,   // (4,1024,64,64)
    const float* __restrict__ rois,   // (N,5)
    float* __restrict__ out,          // (N,1024,7,7)
    int nrois)
{
    const int lane   = threadIdx.x & 31;
    const int lane15 = lane & 15;
    const bool hi    = lane >= 16;          // bottom corner row (K=2,3)

    // Scalarize the wave id so roi-param loads become s_loads.
    const int wslot = __builtin_amdgcn_readfirstlane((int)(threadIdx.x >> 5));
    const int wid   = (int)blockIdx.x * 8 + wslot;
    const int g     = wid & 3;                 // bin group (16 bins)
    const int t     = (wid >> 2) & (CH_TILES - 1); // channel tile (16 ch)
    const int n     = wid >> 8;                // roi index
    if (n >= nrois) return;

    // ---- roi parameters (wave-uniform) ----
    const float* r = rois + (size_t)n * 5;
    const int   bb = (int)r[0];
    const float x1 = r[1] * SPATIAL_SCALE;
    const float y1 = r[2] * SPATIAL_SCALE;
    const float x2 = r[3] * SPATIAL_SCALE;
    const float y2 = r[4] * SPATIAL_SCALE;
    const float binw = fmaxf(x2 - x1, 0.f) * (1.f / 6.f);
    const float binh = fmaxf(y2 - y1, 0.f) * (1.f / 6.f);

    // ---- per-lane bin setup: lane15 selects the bin within the group ----
    const int bi = g * 16 + lane15;            // 0..63 (>=49 -> dead bin)
    const int ph = bi / 7;
    const int pw = bi - ph * 7;
    const float h  = y1 + (float)ph * binh;
    const float w  = x1 + (float)pw * binw;
    const float hf = fminf(floorf(h), (float)(H_ - 2));
    const float wf = fminf(floorf(w), (float)(W_ - 2));
    const float hr = h - hf;                   // ratio (ref semantics, may exceed 1)
    const float wr = w - wf;
    const int hs = min(max((int)hf, 0), H_ - 2);
    const int ws = min(max((int)wf, 0), W_ - 2);
    const bool valid = (bi < NBINS) && (h >= 0.f) && (h < (float)H_)
                                   && (w >= 0.f) && (w < (float)W_);

    // Lane halves carry the two corner rows:
    //  lanes 0-15 : (K0,K1) = (UL,UR) weights, top feature row
    //  lanes16-31 : (K2,K3) = (DL,DR) weights, bottom feature row
    const float u  = hi ? hr : (1.f - hr);
    const float wA = valid ? u * (1.f - wr) : 0.f;  // K0 / K2
    const float wB = valid ? u * wr         : 0.f;  // K1 / K3
    const int   so = hs * W_ + ws;             // spatial offset of bin's UL corner

    // Per-lane gather base: batch + channel (= lane15 of the tile) + row half.
    const float* fp = feat
        + ((size_t)bb * C_ + (size_t)(t * 16 + lane15)) * HW_
        + (hi ? W_ : 0);

    v8f acc = {};  // C/D 16x16 f32 accumulator (8 VGPRs)

    // 16 chained WMMAs: WMMA j contributes only row j (bin g*16+j) of D.
    // B_j = 4x16 corner features of bin j (one float-pair load per lane),
    // A_j = weights of bin j placed in row j (zero elsewhere).
#pragma unroll
    for (int j = 0; j < 16; ++j) {
        const int soj = __shfl(so, j, 32);     // broadcast bin j's corner offset
        const float* p = fp + soj;
        v2f b;
        b.x = p[0];                            // K0 (UL) / K2 (DL)
        b.y = p[1];                            // K1 (UR) / K3 (DR)
        const bool sel = (lane15 == j);
        v2f a;
        a.x = sel ? wA : 0.f;
        a.y = sel ? wB : 0.f;
        acc = __builtin_amdgcn_wmma_f32_16x16x4_f32(
            /*neg_a=*/false, a, /*neg_b=*/false, b,
            /*c_mod=*/(short)0, acc, /*reuse_a=*/false, /*reuse_b=*/false);
    }

    // ---- store: lane = channel, VGPR r = bin (contiguous innermost dim) ----
    // D layout: lanes 0-15 -> bins m = g*16 + r ; lanes 16-31 -> m = g*16+8+r.
    const int mbase = g * 16 + (hi ? 8 : 0);
    float* op = out + ((size_t)n * C_ + (size_t)(t * 16 + lane15)) * NBINS + mbase;

    if (g < 3) {                               // all 16 bins valid
#pragma unroll
        for (int rr = 0; rr < 8; ++rr)
            __builtin_nontemporal_store(acc[rr], op + rr);
    } else if (!hi) {                          // group 3: only bin 48
        __builtin_nontemporal_store(acc[0], op);
    }
}

extern "C" void kernel_launch(void* const* d_in, const int* in_sizes, int n_in,
                              void* d_out, int out_size, void* d_ws, size_t ws_size,
                              hipStream_t stream) {
    const float* feat = (const float*)d_in[0];
    const float* rois = (const float*)d_in[1];
    float* out        = (float*)d_out;

    const int nrois = in_sizes[1] / 5;                 // 2048
    const int waves = nrois * WAVES_PER_ROI;           // 524288
    const int blocks = (waves + 7) / 8;                // 8 waves / 256-thread block

    hipLaunchKernelGGL(roialign_wmma_kernel, dim3(blocks), dim3(256), 0, stream,
                       feat, rois, out, nrois);
}